// DiverseGINE_9225589751990
// MI455X (gfx1250) — compile-verified
//
#include <hip/hip_runtime.h>
#include <math.h>

// ---------------- problem constants ----------------
#define NNODE 200
#define NFEAT 200
#define HID   64
#define EMBD  128
#define BG    64
#define NT    12800      // total nodes
#define NE    819200     // total edges
#define FLATD 13000

typedef float v2f __attribute__((ext_vector_type(2)));
typedef float v8f __attribute__((ext_vector_type(8)));

__device__ __forceinline__ v8f wmma4(v2f a, v2f b, v8f c) {
  // D = A(16x4,f32) * B(4x16,f32) + C(16x16,f32)
  return __builtin_amdgcn_wmma_f32_16x16x4_f32(false, a, false, b, (short)0, c,
                                               false, false);
}

__device__ __forceinline__ unsigned short f2bf(float x) {
  unsigned u = __float_as_uint(x);
  u += 0x7fffu + ((u >> 16) & 1u);       // round-to-nearest-even
  return (unsigned short)(u >> 16);
}
__device__ __forceinline__ float bf2f(unsigned short h) {
  return __uint_as_float(((unsigned)h) << 16);
}

struct SelState { unsigned prefix; unsigned k; };
struct Stats    { double sum; double sumsq; unsigned cnt; unsigned pad;
                  float thr; float mean; float scale; float pad2; };

// ---------------- workspace layout (bytes) ----------------
static constexpr size_t SZ_EMB   = (size_t)NE * HID * 2;   // bf16 edge_emb
static constexpr size_t OFF_EMB  = 0;
static constexpr size_t OFF_DELTA= OFF_EMB  + SZ_EMB;
static constexpr size_t OFF_STR  = OFF_DELTA+ (size_t)NE * 4;
static constexpr size_t OFF_H0   = OFF_STR  + (size_t)NT * 4;
static constexpr size_t OFF_H1   = OFF_H0   + (size_t)NT * HID * 4;
static constexpr size_t OFF_TA   = OFF_H1   + (size_t)NT * HID * 4;
static constexpr size_t OFF_TB   = OFF_TA   + (size_t)NT * HID * 4;
static constexpr size_t OFF_AGG  = OFF_TB   + (size_t)NT * HID * 4;
static constexpr size_t OFF_Y1   = OFF_AGG  + (size_t)NT * HID * 4;
static constexpr size_t OFF_Y1B  = OFF_Y1   + (size_t)BG * 512 * 4;
static constexpr size_t OFF_Y2   = OFF_Y1B  + (size_t)BG * 512 * 4;
static constexpr size_t OFF_Y2B  = OFF_Y2   + (size_t)BG * 256 * 4;
static constexpr size_t OFF_HIST = OFF_Y2B  + (size_t)BG * 256 * 4;
static constexpr size_t OFF_SEL  = OFF_HIST + 256 * 4;
static constexpr size_t OFF_STAT = OFF_SEL  + 64;

// =================== kth-value radix select ===================
__global__ void k_init(const float* __restrict__ kr, SelState* sel, Stats* st,
                       unsigned* hist) {
  int t = threadIdx.x;
  hist[t] = 0u;
  if (t == 0) {
    long long k = (long long)((double)NE * (double)kr[0]);
    if (k < 1) k = 1;
    sel->prefix = 0u; sel->k = (unsigned)k;
    st->sum = 0.0; st->sumsq = 0.0; st->cnt = 0u;
    st->thr = 0.f; st->mean = 0.f; st->scale = 0.f;
  }
}

__global__ void k_hist(const float* __restrict__ ea, const SelState* __restrict__ sel,
                       unsigned* __restrict__ hist, int pass) {
  __shared__ unsigned lh[256];
  int t = threadIdx.x;
  lh[t] = 0u;
  __syncthreads();
  unsigned prefix = sel->prefix;
  int shift = 24 - 8 * pass;
  for (int i = blockIdx.x * blockDim.x + t; i < NE; i += gridDim.x * blockDim.x) {
    unsigned bits = __float_as_uint(ea[i]) & 0x7fffffffu;   // |ea| bit pattern
    bool ok = (pass == 0) || ((bits >> (shift + 8)) == prefix);
    if (ok) atomicAdd(&lh[(bits >> shift) & 255u], 1u);
  }
  __syncthreads();
  if (lh[t]) atomicAdd(&hist[t], lh[t]);
}

__global__ void k_select(SelState* sel, unsigned* hist, Stats* st, int pass) {
  __shared__ unsigned h[256];
  int t = threadIdx.x;
  h[t] = hist[t];
  __syncthreads();
  if (t == 0) {
    unsigned k = sel->k, cum = 0u;
    int b = 0;
    for (int i = 255; i >= 0; --i) {
      unsigned c = h[i];
      if (cum + c >= k) { b = i; break; }
      cum += c;
    }
    sel->k = k - cum;
    unsigned pfx = (sel->prefix << 8) | (unsigned)b;
    sel->prefix = pfx;
    if (pass == 3) st->thr = __uint_as_float(pfx); // exact k-th largest |ea|
  }
  __syncthreads();
  hist[t] = 0u;                                    // ready for next pass
}

// =================== edge stats + node strength ===================
__global__ void k_stats(const float* __restrict__ ea, const int* __restrict__ ei,
                        Stats* st, float* __restrict__ strength) {
  int i = blockIdx.x * blockDim.x + threadIdx.x;
  float thr = st->thr;
  float v = 0.f; unsigned c = 0u;
  float e = ea[i];
  if (fabsf(e) >= thr) {
    v = e; c = 1u;
    atomicAdd(&strength[ei[NE + i]], fabsf(e));   // dst scatter of |sp|
  }
  double s = (double)v, q = (double)v * (double)v;
  for (int m = 16; m; m >>= 1) {
    s += __shfl_xor(s, m, 32);
    q += __shfl_xor(q, m, 32);
    c += __shfl_xor(c, m, 32);
  }
  if ((threadIdx.x & 31) == 0) {
    atomicAdd(&st->sum, s);
    atomicAdd(&st->sumsq, q);
    atomicAdd(&st->cnt, c);
  }
}

__global__ void k_fin(Stats* st) {
  double cnt = (double)st->cnt;
  double mean = st->sum / cnt;
  double var = (st->sumsq - st->sum * st->sum / cnt) / (cnt - 1.0);
  if (var < 0.0) var = 0.0;
  st->mean = (float)mean;
  st->scale = (float)(2.0 / (sqrt(var) + 1e-6));
}

__global__ void k_delta(const float* __restrict__ ea, const Stats* __restrict__ st,
                        float* __restrict__ delta) {
  int i = blockIdx.x * blockDim.x + threadIdx.x;
  float e = ea[i];
  delta[i] = (fabsf(e) >= st->thr) ? (e - st->mean) * st->scale : 0.f;
}

// =================== input GEMM: h = concat @ W_in + b_in ===================
// concat = [x | one-hot(node%200) | strength] -> GEMM over K=200 + lookup + rank1
__global__ void k_hin(const float* __restrict__ x, const float* __restrict__ Win,
                      const float* __restrict__ bin, const float* __restrict__ strength,
                      float* __restrict__ h) {
  int wave = (blockIdx.x * blockDim.x + threadIdx.x) >> 5;
  int lane = threadIdx.x & 31, half = lane >> 4, lr = lane & 15;
  int tm = wave >> 2, tn = wave & 3;               // 800 x 4 tiles
  int row0 = tm * 16, col0 = tn * 16;
  v8f acc = {};
  for (int k0 = 0; k0 < NFEAT; k0 += 4) {
    int ka = k0 + half * 2;
    const float* xp = x + (size_t)(row0 + lr) * NFEAT + ka;
    v2f a; a.x = xp[0]; a.y = xp[1];
    v2f b; b.x = Win[ka * HID + col0 + lr];
           b.y = Win[(ka + 1) * HID + col0 + lr];
    acc = wmma4(a, b, acc);
  }
  int col = col0 + lr;
#pragma unroll
  for (int v = 0; v < 8; ++v) {
    int row = row0 + v + half * 8;
    float val = acc[v]
              + Win[(NFEAT + (row % NNODE)) * HID + col]   // one-hot block
              + strength[row] * Win[400 * HID + col]       // strength column
              + bin[col];
    h[(size_t)row * HID + col] = val;
  }
}

// ========= edge MLP: emb = relu(delta@We1+be1)@We2 + be2 (store bf16) =========
__global__ void k_edge_emb(const float* __restrict__ delta, const float* __restrict__ We1,
                           const float* __restrict__ be1, const float* __restrict__ We2,
                           const float* __restrict__ be2, unsigned short* __restrict__ emb) {
  int wave = (blockIdx.x * blockDim.x + threadIdx.x) >> 5;
  int lane = threadIdx.x & 31, half = lane >> 4, lr = lane & 15;
  int e0 = wave * 16;
  float d = delta[e0 + lr];
  v8f a0 = {}, a1 = {}, a2 = {}, a3 = {};
  for (int k0 = 0; k0 < HID; k0 += 4) {
    int ka = k0 + half * 2;
    v2f a;
    a.x = fmaxf(fmaf(d, We1[ka],     be1[ka]),     0.f);
    a.y = fmaxf(fmaf(d, We1[ka + 1], be1[ka + 1]), 0.f);
    const float* w2 = We2 + ka * HID + lr;
    v2f b;
    b.x = w2[0];  b.y = w2[HID];      a0 = wmma4(a, b, a0);
    b.x = w2[16]; b.y = w2[HID + 16]; a1 = wmma4(a, b, a1);
    b.x = w2[32]; b.y = w2[HID + 32]; a2 = wmma4(a, b, a2);
    b.x = w2[48]; b.y = w2[HID + 48]; a3 = wmma4(a, b, a3);
  }
#pragma unroll
  for (int v = 0; v < 8; ++v) {
    size_t base = (size_t)(e0 + v + half * 8) * HID + lr;
    emb[base]      = f2bf(a0[v] + be2[lr]);
    emb[base + 16] = f2bf(a1[v] + be2[lr + 16]);
    emb[base + 32] = f2bf(a2[v] + be2[lr + 32]);
    emb[base + 48] = f2bf(a3[v] + be2[lr + 48]);
  }
}

// ============ GINE message: agg[dst] += relu(h[src] + emb) ============
__global__ void k_gather(const float* __restrict__ h, const unsigned short* __restrict__ emb,
                         const int* __restrict__ ei, float* __restrict__ agg) {
  int tid = blockIdx.x * blockDim.x + threadIdx.x;   // NE*16 threads
  int e = tid >> 4, f = (tid & 15) * 4;
  int src = ei[e], dst = ei[NE + e];
  const float4 hv = *(const float4*)(h + (size_t)src * HID + f);
  unsigned long long ev = *(const unsigned long long*)(emb + (size_t)e * HID + f);
  float m0 = fmaxf(hv.x + bf2f((unsigned short)(ev)),       0.f);
  float m1 = fmaxf(hv.y + bf2f((unsigned short)(ev >> 16)), 0.f);
  float m2 = fmaxf(hv.z + bf2f((unsigned short)(ev >> 32)), 0.f);
  float m3 = fmaxf(hv.w + bf2f((unsigned short)(ev >> 48)), 0.f);
  float* ap = agg + (size_t)dst * HID + f;
  atomicAdd(ap + 0, m0); atomicAdd(ap + 1, m1);
  atomicAdd(ap + 2, m2); atomicAdd(ap + 3, m3);
}

// ============ GINE MLP part 1: tA = ((1+eps)h + agg) @ Wa + ba ============
__global__ void k_gmlp1(const float* __restrict__ h, const float* __restrict__ agg,
                        const float* __restrict__ epsp, const float* __restrict__ Wa,
                        const float* __restrict__ ba, float* __restrict__ out) {
  int wave = (blockIdx.x * blockDim.x + threadIdx.x) >> 5;
  int lane = threadIdx.x & 31, half = lane >> 4, lr = lane & 15;
  int tm = wave >> 2, tn = wave & 3;
  int row0 = tm * 16, col0 = tn * 16;
  float ep1 = 1.f + epsp[0];
  v8f acc = {};
  for (int k0 = 0; k0 < HID; k0 += 4) {
    int ka = k0 + half * 2;
    size_t ri = (size_t)(row0 + lr) * HID + ka;
    v2f a;
    a.x = fmaf(ep1, h[ri],     agg[ri]);
    a.y = fmaf(ep1, h[ri + 1], agg[ri + 1]);
    v2f b; b.x = Wa[ka * HID + col0 + lr]; b.y = Wa[(ka + 1) * HID + col0 + lr];
    acc = wmma4(a, b, acc);
  }
  int col = col0 + lr;
#pragma unroll
  for (int v = 0; v < 8; ++v) {
    int row = row0 + v + half * 8;
    out[(size_t)row * HID + col] = acc[v] + ba[col];
  }
}

// ============ layernorm(64) + relu, one wave per row ============
__global__ void k_ln64(const float* __restrict__ in, const float* __restrict__ g,
                       const float* __restrict__ b, float* __restrict__ out) {
  int r = (blockIdx.x * blockDim.x + threadIdx.x) >> 5;
  int lane = threadIdx.x & 31;
  float x0 = in[(size_t)r * 64 + lane], x1 = in[(size_t)r * 64 + lane + 32];
  float s = x0 + x1, q = x0 * x0 + x1 * x1;
  for (int m = 16; m; m >>= 1) { s += __shfl_xor(s, m, 32); q += __shfl_xor(q, m, 32); }
  float mu = s * (1.f / 64.f);
  float var = q * (1.f / 64.f) - mu * mu;
  float rs = rsqrtf(var + 1e-5f);
  out[(size_t)r * 64 + lane]      = fmaxf((x0 - mu) * rs * g[lane] + b[lane], 0.f);
  out[(size_t)r * 64 + lane + 32] = fmaxf((x1 - mu) * rs * g[lane + 32] + b[lane + 32], 0.f);
}

// ============ GINE MLP part 2: h' = relu(tB @ Wb + bb) ============
__global__ void k_gmlp2(const float* __restrict__ in, const float* __restrict__ Wb,
                        const float* __restrict__ bb, float* __restrict__ out) {
  int wave = (blockIdx.x * blockDim.x + threadIdx.x) >> 5;
  int lane = threadIdx.x & 31, half = lane >> 4, lr = lane & 15;
  int tm = wave >> 2, tn = wave & 3;
  int row0 = tm * 16, col0 = tn * 16;
  v8f acc = {};
  for (int k0 = 0; k0 < HID; k0 += 4) {
    int ka = k0 + half * 2;
    const float* ip = in + (size_t)(row0 + lr) * HID + ka;
    v2f a; a.x = ip[0]; a.y = ip[1];
    v2f b; b.x = Wb[ka * HID + col0 + lr]; b.y = Wb[(ka + 1) * HID + col0 + lr];
    acc = wmma4(a, b, acc);
  }
  int col = col0 + lr;
#pragma unroll
  for (int v = 0; v < 8; ++v) {
    int row = row0 + v + half * 8;
    out[(size_t)row * HID + col] = fmaxf(acc[v] + bb[col], 0.f);
  }
}

// ============ head GEMM1 (split-K): y1 += flat @ Wh1 ============
// flat[r][k] = k<12800 ? h[r*12800+k] : strength[r*200 + k-12800]
__global__ void k_head1(const float* __restrict__ h, const float* __restrict__ strength,
                        const float* __restrict__ Wh1, float* __restrict__ y1) {
  int wave = (blockIdx.x * blockDim.x + threadIdx.x) >> 5;  // 1280 waves
  int lane = threadIdx.x & 31, half = lane >> 4, lr = lane & 15;
  int chunk = wave / 128, t = wave % 128;
  int tm = t >> 5, tn = t & 31;                              // 4 x 32 tiles
  int row0 = tm * 16, col0 = tn * 16;
  int kbeg = chunk * 1300, kend = kbeg + 1300;
  int r = row0 + lr;
  v8f acc = {};
  for (int k0 = kbeg; k0 < kend; k0 += 4) {
    int ka = k0 + half * 2;
    v2f a;
    a.x = (ka     < 12800) ? h[(size_t)r * 12800 + ka]     : strength[r * 200 + ka - 12800];
    a.y = (ka + 1 < 12800) ? h[(size_t)r * 12800 + ka + 1] : strength[r * 200 + ka + 1 - 12800];
    v2f b; b.x = Wh1[(size_t)ka * 512 + col0 + lr];
           b.y = Wh1[(size_t)(ka + 1) * 512 + col0 + lr];
    acc = wmma4(a, b, acc);
  }
  int col = col0 + lr;
#pragma unroll
  for (int v = 0; v < 8; ++v) {
    int row = row0 + v + half * 8;
    atomicAdd(&y1[(size_t)row * 512 + col], acc[v]);
  }
}

// ============ wide layernorm + relu: block per row ============
__global__ void k_lnw(const float* __restrict__ y, const float* __restrict__ bias,
                      const float* __restrict__ g, const float* __restrict__ b,
                      float* __restrict__ out, int ncols) {
  __shared__ float sh[512];
  int r = blockIdx.x, c = threadIdx.x;
  float v = y[(size_t)r * ncols + c] + bias[c];
  sh[c] = v; __syncthreads();
  for (int s = ncols >> 1; s; s >>= 1) { if (c < s) sh[c] += sh[c + s]; __syncthreads(); }
  float mu = sh[0] / (float)ncols;
  __syncthreads();
  float d = v - mu;
  sh[c] = d * d; __syncthreads();
  for (int s = ncols >> 1; s; s >>= 1) { if (c < s) sh[c] += sh[c + s]; __syncthreads(); }
  float rs = rsqrtf(sh[0] / (float)ncols + 1e-5f);
  out[(size_t)r * ncols + c] = fmaxf(d * rs * g[c] + b[c], 0.f);
}

// ============ head GEMM2: y2 = y1b @ Wh2 (bias in LN kernel) ============
__global__ void k_head2(const float* __restrict__ in, const float* __restrict__ Wh2,
                        float* __restrict__ out) {
  int wave = (blockIdx.x * blockDim.x + threadIdx.x) >> 5;   // 64 waves
  int lane = threadIdx.x & 31, half = lane >> 4, lr = lane & 15;
  int tm = wave >> 4, tn = wave & 15;                        // 4 x 16 tiles
  int row0 = tm * 16, col0 = tn * 16;
  v8f acc = {};
  for (int k0 = 0; k0 < 512; k0 += 4) {
    int ka = k0 + half * 2;
    const float* ip = in + (size_t)(row0 + lr) * 512 + ka;
    v2f a; a.x = ip[0]; a.y = ip[1];
    v2f b; b.x = Wh2[ka * 256 + col0 + lr]; b.y = Wh2[(ka + 1) * 256 + col0 + lr];
    acc = wmma4(a, b, acc);
  }
  int col = col0 + lr;
#pragma unroll
  for (int v = 0; v < 8; ++v)
    out[(size_t)(row0 + v + half * 8) * 256 + col] = acc[v];
}

// ============ head GEMM3: out = y2b @ Wh3 + bh3 ============
__global__ void k_head3(const float* __restrict__ in, const float* __restrict__ Wh3,
                        const float* __restrict__ bh3, float* __restrict__ out) {
  int wave = (blockIdx.x * blockDim.x + threadIdx.x) >> 5;   // 32 waves
  int lane = threadIdx.x & 31, half = lane >> 4, lr = lane & 15;
  int tm = wave >> 3, tn = wave & 7;                         // 4 x 8 tiles
  int row0 = tm * 16, col0 = tn * 16;
  v8f acc = {};
  for (int k0 = 0; k0 < 256; k0 += 4) {
    int ka = k0 + half * 2;
    const float* ip = in + (size_t)(row0 + lr) * 256 + ka;
    v2f a; a.x = ip[0]; a.y = ip[1];
    v2f b; b.x = Wh3[ka * EMBD + col0 + lr]; b.y = Wh3[(ka + 1) * EMBD + col0 + lr];
    acc = wmma4(a, b, acc);
  }
  int col = col0 + lr;
#pragma unroll
  for (int v = 0; v < 8; ++v)
    out[(size_t)(row0 + v + half * 8) * EMBD + col] = acc[v] + bh3[col];
}

// =========================== launch ===========================
extern "C" void kernel_launch(void* const* d_in, const int* in_sizes, int n_in,
                              void* d_out, int out_size, void* d_ws, size_t ws_size,
                              hipStream_t stream) {
  const float* x    = (const float*)d_in[0];
  const int*   ei   = (const int*)  d_in[1];
  const float* ea   = (const float*)d_in[2];
  const float* keep = (const float*)d_in[4];
  // params flattened in sorted-key (jax pytree) order:
  const float* W0a  = (const float*)d_in[5];
  const float* W0b  = (const float*)d_in[6];
  const float* W1a  = (const float*)d_in[7];
  const float* W1b  = (const float*)d_in[8];
  const float* Win  = (const float*)d_in[9];
  const float* We1  = (const float*)d_in[10];
  const float* We2  = (const float*)d_in[11];
  const float* Wh1  = (const float*)d_in[12];
  const float* Wh2  = (const float*)d_in[13];
  const float* Wh3  = (const float*)d_in[14];
  const float* b0a  = (const float*)d_in[15];
  const float* b0b  = (const float*)d_in[16];
  const float* b1a  = (const float*)d_in[17];
  const float* b1b  = (const float*)d_in[18];
  const float* b_h1 = (const float*)d_in[19];
  const float* b_h2 = (const float*)d_in[20];
  const float* b_in = (const float*)d_in[21];
  const float* bb0  = (const float*)d_in[22];
  const float* bb1  = (const float*)d_in[23];
  const float* be1  = (const float*)d_in[24];
  const float* be2  = (const float*)d_in[25];
  const float* bh1  = (const float*)d_in[26];
  const float* bh2  = (const float*)d_in[27];
  const float* bh3  = (const float*)d_in[28];
  const float* eps0 = (const float*)d_in[29];
  const float* eps1 = (const float*)d_in[30];
  const float* g0   = (const float*)d_in[31];
  const float* g1   = (const float*)d_in[32];
  const float* g_h1 = (const float*)d_in[33];
  const float* g_h2 = (const float*)d_in[34];

  char* ws = (char*)d_ws;
  unsigned short* emb = (unsigned short*)(ws + OFF_EMB);
  float* delta    = (float*)(ws + OFF_DELTA);
  float* strength = (float*)(ws + OFF_STR);
  float* h0       = (float*)(ws + OFF_H0);
  float* h1       = (float*)(ws + OFF_H1);
  float* tA       = (float*)(ws + OFF_TA);
  float* tB       = (float*)(ws + OFF_TB);
  float* agg      = (float*)(ws + OFF_AGG);
  float* y1       = (float*)(ws + OFF_Y1);
  float* y1b      = (float*)(ws + OFF_Y1B);
  float* y2       = (float*)(ws + OFF_Y2);
  float* y2b      = (float*)(ws + OFF_Y2B);
  unsigned* hist  = (unsigned*)(ws + OFF_HIST);
  SelState* sel   = (SelState*)(ws + OFF_SEL);
  Stats* st       = (Stats*)(ws + OFF_STAT);
  float* outp     = (float*)d_out;

  hipMemsetAsync(strength, 0, (size_t)NT * 4, stream);
  k_init<<<1, 256, 0, stream>>>(keep, sel, st, hist);

  for (int p = 0; p < 4; ++p) {
    k_hist<<<512, 256, 0, stream>>>(ea, sel, hist, p);
    k_select<<<1, 256, 0, stream>>>(sel, hist, st, p);
  }
  k_stats<<<NE / 256, 256, 0, stream>>>(ea, ei, st, strength);
  k_fin<<<1, 1, 0, stream>>>(st);
  k_delta<<<NE / 256, 256, 0, stream>>>(ea, st, delta);

  k_hin<<<400, 256, 0, stream>>>(x, Win, b_in, strength, h0);           // 3200 waves
  k_edge_emb<<<6400, 256, 0, stream>>>(delta, We1, be1, We2, be2, emb); // 51200 waves

  // ---- GINE layer 0 ----
  hipMemsetAsync(agg, 0, (size_t)NT * HID * 4, stream);
  k_gather<<<NE * 16 / 256, 256, 0, stream>>>(h0, emb, ei, agg);
  k_gmlp1<<<400, 256, 0, stream>>>(h0, agg, eps0, W0a, b0a, tA);
  k_ln64<<<NT * 32 / 256, 256, 0, stream>>>(tA, g0, bb0, tB);
  k_gmlp2<<<400, 256, 0, stream>>>(tB, W0b, b0b, h1);
  // ---- GINE layer 1 ----
  hipMemsetAsync(agg, 0, (size_t)NT * HID * 4, stream);
  k_gather<<<NE * 16 / 256, 256, 0, stream>>>(h1, emb, ei, agg);
  k_gmlp1<<<400, 256, 0, stream>>>(h1, agg, eps1, W1a, b1a, tA);
  k_ln64<<<NT * 32 / 256, 256, 0, stream>>>(tA, g1, bb1, tB);
  k_gmlp2<<<400, 256, 0, stream>>>(tB, W1b, b1b, h0);

  // ---- head ----
  hipMemsetAsync(y1, 0, (size_t)BG * 512 * 4, stream);
  k_head1<<<160, 256, 0, stream>>>(h0, strength, Wh1, y1);   // 1280 waves, split-K
  k_lnw<<<BG, 512, 0, stream>>>(y1, bh1, g_h1, b_h1, y1b, 512);
  k_head2<<<8, 256, 0, stream>>>(y1b, Wh2, y2);
  k_lnw<<<BG, 256, 0, stream>>>(y2, bh2, g_h2, b_h2, y2b, 256);
  k_head3<<<4, 256, 0, stream>>>(y2b, Wh3, bh3, outp);
}